// TridiagonalResolventLayer_1382979470227
// MI455X (gfx1250) — compile-verified
//
#include <hip/hip_runtime.h>
#include <hip/hip_bf16.h>
#include <math.h>

// ---------------------------------------------------------------------------
// y = x @ Minv^T where Minv is the (banded, fp32-exact) inverse of the fixed
// tridiagonal M.  Diagonal dominance => |Minv[i,j]| decays ~0.16^|i-j|, so the
// block-tridiagonal band (coverage |i-j| in [16,31]) is exact far below fp32
// eps.  One pass over x/y: 512 MB @ 23.3 TB/s ~= 22 us floor.
// Compute: 12 V_WMMA_F32_16X16X4_F32 per 16x16 output tile (~6.4 GFLOP total).
// This round: fully branch-free inner loop via zero-padded LDS columns and
// zero-filled edge W slices (no EXEC masking around WMMA).
// ---------------------------------------------------------------------------

#define N_FEAT   2048
#define NB       128          // feature blocks of 16
#define REL      3            // band block-diagonals: Jb = Ib-1, Ib, Ib+1
#define WIN      40           // half-window for local Thomas solve
#define LDS_ROW  2084         // padded row stride (floats): 16 left pad +
                              // 2048 data + 20 right pad; 2084%64==36 keeps
                              // the 16x4 fragment reads on all 64 banks
#define LPAD     16           // left zero-pad columns
#define ROWS_PER_BLOCK 16

typedef __attribute__((ext_vector_type(2))) float v2f;
typedef __attribute__((ext_vector_type(8))) float v8f;

// ---------------------------------------------------------------------------
// Kernel 1: build banded inverse, pre-swizzled into WMMA B-fragment order.
// One thread per column j: solve M_win yw = e_j on window [j-WIN, j+WIN],
// then scatter so that in the main kernel lane L reads a contiguous float2:
//   W[(((Ib*3+t)*4 + kc)*64) + lane*2 + p]  =  Minv[16*Ib + n, 16*Jb + j']
// with Jb = Ib-1+t, j' = 4*kc + 2*half + p, lane = 16*half + n.
// The two slices never inside the band (Ib=0,t=0 / Ib=127,t=2) are zeroed so
// the main kernel needs no edge branches.
// ---------------------------------------------------------------------------
__global__ __launch_bounds__(256)
void build_band_inverse(const float* __restrict__ dp,
                        const float* __restrict__ lp,
                        const float* __restrict__ up,
                        float* __restrict__ W) {
    int j = blockIdx.x * blockDim.x + threadIdx.x;
    if (j >= N_FEAT) return;

    // zero the two edge slices (256 floats each) so main kernel is branch-free
    if (j == 0) {
        for (int k = 0; k < 256; ++k) W[k] = 0.0f;                   // (Ib=0,t=0)
    }
    if (j == N_FEAT - 1) {
        size_t base = (size_t)((NB - 1) * REL + 2) * 256;
        for (int k = 0; k < 256; ++k) W[base + k] = 0.0f;            // (Ib=127,t=2)
    }

    int lo = j - WIN; if (lo < 0) lo = 0;
    int hi = j + WIN; if (hi > N_FEAT - 1) hi = N_FEAT - 1;
    int m  = hi - lo + 1;                       // <= 81

    float cp[2*WIN + 1];
    float yw[2*WIN + 1];

    // Thomas forward sweep on the window, RHS = e_j
    float prev_c = 0.0f, prev_z = 0.0f;
    for (int t = 0; t < m; ++t) {
        int row = lo + t;
        float dv  = dp[row];
        float sp  = (dv > 20.0f) ? dv : log1pf(expf(dv));   // softplus
        float a   = sp + 2.0f;                              // diag
        float sub = (t > 0)            ? tanhf(lp[row - 1]) : 0.0f;
        float sup = (row < N_FEAT - 1) ? tanhf(up[row])     : 0.0f;
        float rhs = (row == j) ? 1.0f : 0.0f;
        float inv = 1.0f / (a - sub * prev_c);
        float c   = sup * inv;
        float z   = (rhs - sub * prev_z) * inv;
        cp[t] = c;  yw[t] = z;
        prev_c = c; prev_z = z;
    }
    // back substitution in place
    for (int t = m - 2; t >= 0; --t) yw[t] -= cp[t] * yw[t + 1];

    // scatter band entries into fragment-friendly layout
    int Jb = j >> 4, jq = j & 15;
    int kc = jq >> 2, rem = jq & 3, hf = rem >> 1, p = rem & 1;
    for (int t = 0; t < REL; ++t) {
        int Ib = Jb + 1 - t;                    // t = Jb - Ib + 1
        if (Ib < 0 || Ib >= NB) continue;
        for (int n = 0; n < 16; ++n) {
            int i = Ib * 16 + n;                // always inside [lo,hi]
            float v = (i >= lo && i <= hi) ? yw[i - lo] : 0.0f;
            int lane = hf * 16 + n;
            W[(size_t)(((Ib * REL + t) * 4 + kc) * 64) + lane * 2 + p] = v;
        }
    }
}

// ---------------------------------------------------------------------------
// Kernel 2: banded block-GEMM.  Block = 256 threads (8 waves) owns 16 rows
// of x; zero-fills the pad columns, stages 16x2048 fp32 into LDS with
// global_load_async_to_lds_b128, then wave w computes output tiles for
// feature blocks Ib in [16w, 16w+16) with an unconditional 12-WMMA chain.
// ---------------------------------------------------------------------------
__global__ __launch_bounds__(256)
void banded_apply(const float* __restrict__ x,
                  const float* __restrict__ W,
                  float* __restrict__ y) {
    extern __shared__ float xs[];               // [16][LDS_ROW], data at +LPAD
    const int tid  = threadIdx.x;
    const long s0  = (long)blockIdx.x * ROWS_PER_BLOCK;
    const float* xbase = x + s0 * N_FEAT;

    // ---- zero the pad columns: 16 rows x (16 left + 20 right) floats ----
    for (int e = tid; e < 16 * 36; e += 256) {
        int row = e / 36, idx = e % 36;
        int colidx = (idx < LPAD) ? idx : (LPAD + N_FEAT + idx - LPAD);
        xs[row * LDS_ROW + colidx] = 0.0f;
    }

    // ---- async stage: 8192 x 16B chunks, coalesced, ASYNCcnt tracked ----
    unsigned lds_base = (unsigned)(size_t)(void*)xs;   // low 32b = LDS offset
    #pragma unroll 4
    for (int k = 0; k < 32; ++k) {
        int chunk = k * 256 + tid;              // 512 chunks per row
        int row   = chunk >> 9;
        int c16   = chunk & 511;
        unsigned goff = (unsigned)(row * (N_FEAT * 4) + c16 * 16);
        unsigned loff = lds_base +
                        (unsigned)(row * (LDS_ROW * 4) + LPAD * 4 + c16 * 16);
        asm volatile("global_load_async_to_lds_b128 %0, %1, %2"
                     :: "v"(loff), "v"(goff), "s"(xbase) : "memory");
    }
    asm volatile("s_wait_asynccnt 0" ::: "memory");
    __syncthreads();

    // ---- compute: wave-private sweep over 16 feature blocks ----
    const int wave = tid >> 5;
    const int lane = tid & 31;
    const int hf   = lane >> 4;                 // fragment half
    const int r16  = lane & 15;                 // A: row / B,D: column

    // lane-invariant bases hoisted out of the loop
    const float* arow = xs + r16 * LDS_ROW + LPAD + 2 * hf;  // + col
    float*       yrow = y + (s0 + 8 * hf) * N_FEAT + r16;    // + g*N_FEAT + 16*ib

    for (int ib = wave * 16; ib < wave * 16 + 16; ++ib) {
        v8f acc = {0.f, 0.f, 0.f, 0.f, 0.f, 0.f, 0.f, 0.f};
        const float* wb = W + (size_t)(ib * REL) * 256 + lane * 2;
        const int    cb = (ib - 1) * 16;        // first band block column base
        #pragma unroll
        for (int t = 0; t < REL; ++t) {
            #pragma unroll
            for (int kc = 0; kc < 4; ++kc) {
                // A fragment: 16x4 fp32 (Jb=-1/128 read zero pad columns)
                v2f a = *(const v2f*)(arow + cb + t * 16 + kc * 4);
                // B fragment: pre-swizzled contiguous float2 per lane
                // (edge slices are zero-filled)
                v2f b = *(const v2f*)(wb + (t * 4 + kc) * 64);
                acc = __builtin_amdgcn_wmma_f32_16x16x4_f32(
                          false, a, false, b, (short)0, acc, false, false);
            }
        }
        // D tile store: VGPR g -> row g + 8*hf, column = lane&15
        #pragma unroll
        for (int g = 0; g < 8; ++g) {
            yrow[(long)g * N_FEAT + ib * 16] = acc[g];
        }
    }
}

// ---------------------------------------------------------------------------
extern "C" void kernel_launch(void* const* d_in, const int* in_sizes, int n_in,
                              void* d_out, int out_size, void* d_ws, size_t ws_size,
                              hipStream_t stream) {
    const float* x  = (const float*)d_in[0];   // [32,1024,2048]
    const float* dp = (const float*)d_in[1];   // [2048]
    const float* lp = (const float*)d_in[2];   // [2047]
    const float* up = (const float*)d_in[3];   // [2047]
    float* out = (float*)d_out;
    float* W   = (float*)d_ws;                 // needs 128*3*4*64*4 = 393216 B

    // 1) banded inverse, pre-swizzled for WMMA B fragments (tiny, L2-resident)
    build_band_inverse<<<N_FEAT / 256, 256, 0, stream>>>(dp, lp, up, W);

    // 2) one-pass banded block-GEMM over all rows
    int total_rows = out_size / N_FEAT;        // 32768
    int row_tiles  = total_rows / ROWS_PER_BLOCK;
    size_t lds_bytes = (size_t)ROWS_PER_BLOCK * LDS_ROW * sizeof(float);
    banded_apply<<<row_tiles, 256, lds_bytes, stream>>>(x, W, out);
}